// constrained_network_77103252897806
// MI455X (gfx1250) — compile-verified
//
#include <hip/hip_runtime.h>
#include <math.h>

#define NN 20000
#define EE 320000
#define NSC 64
#define NVC 32
#define NB 8
#define RH 64
#define EMBD 8
#define FEAT 160         // NSC + 3*NVC
#define MAXR 3.5f
#define HSTEP 0.1f
#define EPSR 1e-9f
#define INV_SQRT_NEIGH 0.25f
#define PI_F 3.14159265358979323846f

typedef __attribute__((ext_vector_type(2))) float v2f;
typedef __attribute__((ext_vector_type(8))) float v8f;

#ifndef __has_builtin
#define __has_builtin(x) 0
#endif

#if __has_builtin(__builtin_amdgcn_global_load_async_to_lds_b128)
#define HAVE_ASYNC_LDS 1
#else
#define HAVE_ASYNC_LDS 0
#endif

typedef int v4i_ __attribute__((vector_size(16)));
typedef __attribute__((address_space(1))) v4i_ as1_v4i;
typedef __attribute__((address_space(3))) v4i_ as3_v4i;

__device__ inline void wait_async0() {
#if __has_builtin(__builtin_amdgcn_s_wait_asynccnt)
    __builtin_amdgcn_s_wait_asynccnt(0);
#else
    asm volatile("s_wait_asynccnt 0x0" ::: "memory");
#endif
}

__device__ inline v8f wmma4(v2f a, v2f b, v8f c) {
    return __builtin_amdgcn_wmma_f32_16x16x4_f32(false, a, false, b, (short)0, c, false, false);
}
__device__ inline float sigmoidf_(float x) { return 1.0f / (1.0f + __expf(-x)); }

// ---------------------------------------------------------------------------
// Generic fp32 WMMA GEMM: C[M,N] = op(A[M,K] @ B[K,N]); one wave per 16x16 tile
// ---------------------------------------------------------------------------
#define GOP_STORE 0
#define GOP_ACC 1
#define GOP_BIAS_SILU 2

__global__ __launch_bounds__(256) void gemm_wmma_f32(
    const float* __restrict__ A, int lda,
    const float* __restrict__ B, int ldb,
    float* __restrict__ C, int ldc,
    int K, const float* __restrict__ bias, int mode,
    int nTiles, int tilesN)
{
    int wid = blockIdx.x * (blockDim.x >> 5) + (threadIdx.x >> 5);
    if (wid >= nTiles) return;
    int tM = wid / tilesN, tN = wid % tilesN;
    int lane = threadIdx.x & 31;
    int row  = lane & 15;
    int half = lane >> 4;
    int col  = lane & 15;

    v8f c = {0.f, 0.f, 0.f, 0.f, 0.f, 0.f, 0.f, 0.f};
    if (mode == GOP_ACC) {
        for (int r = 0; r < 8; ++r)
            c[r] = C[(size_t)(tM * 16 + half * 8 + r) * ldc + tN * 16 + col];
    }
    const float* Ap = A + (size_t)(tM * 16 + row) * lda + half * 2;
    const float* Bp = B + (size_t)(half * 2) * ldb + tN * 16 + col;
    for (int k0 = 0; k0 < K; k0 += 4) {
        v2f a, b;
        a[0] = Ap[k0];
        a[1] = Ap[k0 + 1];
        b[0] = Bp[(size_t)k0 * ldb];
        b[1] = Bp[(size_t)(k0 + 1) * ldb];
        c = wmma4(a, b, c);
    }
    for (int r = 0; r < 8; ++r) {
        int rr = tM * 16 + half * 8 + r;
        int cc = tN * 16 + col;
        float v = c[r];
        if (mode == GOP_BIAS_SILU) { v += bias[cc]; v = v * sigmoidf_(v); }
        C[(size_t)rr * ldc + cc] = v;
    }
}

// ---------------------------------------------------------------------------
// Fused edge kernel: async-stage radial weights into LDS -> geometry ->
// WMMA radial MLP (8->64->128) -> messages -> atomic scatter.
// One wave handles 16 edges; 4 waves per block. Dynamic LDS (~87 KB).
// ---------------------------------------------------------------------------
#define EK_WAVES 4
#define SZ_WR1 (NB * RH)      // 512
#define SZ_BR1 (RH)           // 64
#define SZ_WR2 (RH * 128)     // 8192
#define SZ_EF  (EK_WAVES * 16 * NB)
#define SZ_EA  (EK_WAVES * 16 * 3)
#define SZ_H   (EK_WAVES * 16 * RH)
#define SZ_W   (EK_WAVES * 16 * 128)
#define EK_SMEM_FLOATS (SZ_WR1 + SZ_BR1 + SZ_WR2 + SZ_EF + SZ_EA + SZ_H + SZ_W)

__global__ __launch_bounds__(128) void edge_kernel(
    const int* __restrict__ src, const int* __restrict__ dst,
    const float* __restrict__ xc,
    const float* __restrict__ s1, const float* __restrict__ v1,
    const float* __restrict__ Wr1, const float* __restrict__ br1,
    const float* __restrict__ Wr2,
    float* __restrict__ a0, float* __restrict__ av)
{
    extern __shared__ float smem[];
    float* s_Wr1 = smem;
    float* s_br1 = s_Wr1 + SZ_WR1;
    float* s_Wr2 = s_br1 + SZ_BR1;
    float* s_ef  = s_Wr2 + SZ_WR2;
    float* s_ea  = s_ef + SZ_EF;
    float* s_h   = s_ea + SZ_EA;
    float* s_w   = s_h + SZ_H;

    int wave = threadIdx.x >> 5, lane = threadIdx.x & 31;
    int e0 = (blockIdx.x * EK_WAVES + wave) * 16;

    // ---- stage 0: stage radial-MLP weights into LDS (async DMA path on CDNA5)
#if HAVE_ASYNC_LDS
    for (int i = threadIdx.x; i < SZ_WR1 / 4; i += 128)
        __builtin_amdgcn_global_load_async_to_lds_b128(
            (as1_v4i*)(Wr1 + i * 4), (as3_v4i*)(s_Wr1 + i * 4), 0, 0);
    for (int i = threadIdx.x; i < SZ_BR1 / 4; i += 128)
        __builtin_amdgcn_global_load_async_to_lds_b128(
            (as1_v4i*)(br1 + i * 4), (as3_v4i*)(s_br1 + i * 4), 0, 0);
    for (int i = threadIdx.x; i < SZ_WR2 / 4; i += 128)
        __builtin_amdgcn_global_load_async_to_lds_b128(
            (as1_v4i*)(Wr2 + i * 4), (as3_v4i*)(s_Wr2 + i * 4), 0, 0);
    wait_async0();
    __syncthreads();
#else
    for (int i = threadIdx.x; i < SZ_WR1; i += 128) s_Wr1[i] = Wr1[i];
    for (int i = threadIdx.x; i < SZ_BR1; i += 128) s_br1[i] = br1[i];
    for (int i = threadIdx.x; i < SZ_WR2; i += 128) s_Wr2[i] = Wr2[i];
    __syncthreads();
#endif

    // ---- stage 1: per-edge geometry (bessel basis + cutoff-scaled direction)
    if (lane < 16) {
        int e = e0 + lane;
        int s = src[e], d = dst[e];
        float ev[3];
        float r2 = EPSR;
        for (int i = 0; i < 3; ++i) {
            ev[i] = xc[s * 6 + 3 + i] - xc[d * 6 + 3 + i];
            r2 += ev[i] * ev[i];
        }
        float elen = sqrtf(r2);
        float u = 2.0f * (elen / MAXR - 1.0f);
        float cut = 0.5f * (1.0f - __cosf(PI_F * u));
        cut = (u > 0.0f) ? 0.0f : ((u < -1.0f) ? 1.0f : cut);
        float sc = 1.7320508075688772f / elen * cut; // sqrt(3)/len * cutoff
        for (int i = 0; i < 3; ++i) s_ea[wave * 48 + lane * 3 + i] = sc * ev[i];
        float ds = fmaxf(elen, EPSR);
        const float coef = 0.75592894601845445f * 2.8284271247461903f; // sqrt(2/MAXR)*sqrt(NB)
        for (int n = 1; n <= NB; ++n)
            s_ef[wave * 16 * NB + lane * NB + n - 1] =
                coef * __sinf((float)n * PI_F * ds / MAXR) / ds;
    }
    __syncthreads();

    int row = lane & 15, half = lane >> 4, col = lane & 15;

    // ---- stage 2: h = silu(ef @ Wr1 + br1)   (16x8)@(8x64), B from LDS
    for (int ct = 0; ct < RH / 16; ++ct) {
        v8f c = {0.f, 0.f, 0.f, 0.f, 0.f, 0.f, 0.f, 0.f};
        for (int k0 = 0; k0 < NB; k0 += 4) {
            v2f a, b;
            a[0] = s_ef[wave * 16 * NB + row * NB + k0 + half * 2];
            a[1] = s_ef[wave * 16 * NB + row * NB + k0 + half * 2 + 1];
            b[0] = s_Wr1[(k0 + half * 2) * RH + ct * 16 + col];
            b[1] = s_Wr1[(k0 + half * 2 + 1) * RH + ct * 16 + col];
            c = wmma4(a, b, c);
        }
        for (int r = 0; r < 8; ++r) {
            int cc = ct * 16 + col;
            float v = c[r] + s_br1[cc];
            v = v * sigmoidf_(v);
            s_h[wave * 16 * RH + (half * 8 + r) * RH + cc] = v;
        }
    }
    __syncthreads();

    // ---- stage 3: w = h @ Wr2   (16x64)@(64x128), B from LDS
    for (int ct = 0; ct < 8; ++ct) {
        v8f c = {0.f, 0.f, 0.f, 0.f, 0.f, 0.f, 0.f, 0.f};
        for (int k0 = 0; k0 < RH; k0 += 4) {
            v2f a, b;
            a[0] = s_h[wave * 16 * RH + row * RH + k0 + half * 2];
            a[1] = s_h[wave * 16 * RH + row * RH + k0 + half * 2 + 1];
            b[0] = s_Wr2[(k0 + half * 2) * 128 + ct * 16 + col];
            b[1] = s_Wr2[(k0 + half * 2 + 1) * 128 + ct * 16 + col];
            c = wmma4(a, b, c);
        }
        for (int r = 0; r < 8; ++r)
            s_w[wave * 16 * 128 + (half * 8 + r) * 128 + ct * 16 + col] = c[r];
    }
    __syncthreads();

    // ---- stage 4: messages + atomic scatter (all 32 lanes per edge)
    for (int el = 0; el < 16; ++el) {
        int e = e0 + el;
        int s = src[e], d = dst[e];
        float ea0 = s_ea[wave * 48 + el * 3 + 0];
        float ea1 = s_ea[wave * 48 + el * 3 + 1];
        float ea2 = s_ea[wave * 48 + el * 3 + 2];
        // mva: 64 scalar-gated channels (2 per lane)
        for (int ch = 0; ch < 2; ++ch) {
            int cch = ch * 32 + lane;
            float mv = s_w[wave * 16 * 128 + el * 128 + cch] *
                       s1[(size_t)s * NSC + cch] * INV_SQRT_NEIGH;
            atomicAdd(&av[((size_t)d * 3 + 0) * 96 + cch], mv * ea0);
            atomicAdd(&av[((size_t)d * 3 + 1) * 96 + cch], mv * ea1);
            atomicAdd(&av[((size_t)d * 3 + 2) * 96 + cch], mv * ea2);
        }
        // m0 + mvb: 32 vector channels (1 per lane)
        float vs0 = v1[((size_t)s * 3 + 0) * NVC + lane];
        float vs1 = v1[((size_t)s * 3 + 1) * NVC + lane];
        float vs2 = v1[((size_t)s * 3 + 2) * NVC + lane];
        float wb = s_w[wave * 16 * 128 + el * 128 + 64 + lane];
        atomicAdd(&a0[(size_t)d * NVC + lane],
                  wb * (vs0 * ea0 + vs1 * ea1 + vs2 * ea2) * INV_SQRT_NEIGH);
        float wc = s_w[wave * 16 * 128 + el * 128 + 96 + lane] * INV_SQRT_NEIGH;
        atomicAdd(&av[((size_t)d * 3 + 0) * 96 + 64 + lane], wc * (vs1 * ea2 - vs2 * ea1));
        atomicAdd(&av[((size_t)d * 3 + 1) * 96 + 64 + lane], wc * (vs2 * ea0 - vs0 * ea2));
        atomicAdd(&av[((size_t)d * 3 + 2) * 96 + 64 + lane], wc * (vs0 * ea1 - vs1 * ea0));
    }
}

// ---------------------------------------------------------------------------
// Small helper kernels
// ---------------------------------------------------------------------------
__global__ void semiunitary_kernel(const float* __restrict__ Mp, float* __restrict__ Mo) {
    if (threadIdx.x != 0 || blockIdx.x != 0) return;
    float Mt[2][32], n0[32], n1[32];
    for (int k = 0; k < 2; ++k)
        for (int v = 0; v < 32; ++v) Mt[k][v] = Mp[k * 32 + v];
    for (int it = 0; it < 10; ++it) {
        float g00 = -1.f, g01 = 0.f, g11 = -1.f;
        for (int v = 0; v < 32; ++v) {
            g00 += Mt[0][v] * Mt[0][v];
            g01 += Mt[0][v] * Mt[1][v];
            g11 += Mt[1][v] * Mt[1][v];
        }
        for (int v = 0; v < 32; ++v) {
            n0[v] = Mt[0][v] - 0.5f * (g00 * Mt[0][v] + g01 * Mt[1][v]);
            n1[v] = Mt[1][v] - 0.5f * (g01 * Mt[0][v] + g11 * Mt[1][v]);
        }
        for (int v = 0; v < 32; ++v) { Mt[0][v] = n0[v]; Mt[1][v] = n1[v]; }
    }
    for (int k = 0; k < 2; ++k)
        for (int v = 0; v < 32; ++v) Mo[k * 32 + v] = Mt[k][v];
}

__global__ void zemb_kernel(const int* __restrict__ attr, const float* __restrict__ tab,
                            float* __restrict__ z) {
    int i = blockIdx.x * blockDim.x + threadIdx.x;
    if (i >= NN * EMBD) return;
    int n = i / EMBD, a = i % EMBD;
    z[i] = tab[attr[n] * EMBD + a];
}

__global__ void uplift_kernel(const float* __restrict__ x, const float* __restrict__ M,
                              float* __restrict__ y) {
    int i = blockIdx.x * blockDim.x + threadIdx.x;
    if (i >= NN * FEAT) return;
    int n = i / FEAT, f = i % FEAT;
    if (f < NSC) { y[i] = 0.f; return; }
    int fv = f - NSC, v = fv / 3, c = fv % 3;
    y[i] = x[n * 6 + c] * M[v] + x[n * 6 + 3 + c] * M[NVC + v];
}

__global__ void repack_v_kernel(const float* __restrict__ y, float* __restrict__ vt) {
    int i = blockIdx.x * blockDim.x + threadIdx.x;
    if (i >= NN * 96) return;
    int n = i / 96, rem = i % 96, c = rem / 32, v = rem % 32;
    vt[i] = y[(size_t)n * FEAT + NSC + v * 3 + c]; // vt layout (n,i,v)
}

__global__ void contract_s_kernel(const float* __restrict__ z, const float* __restrict__ t,
                                  float* __restrict__ outs) {
    int i = blockIdx.x * blockDim.x + threadIdx.x;
    if (i >= NN * 96) return;
    int n = i / 96, w = i % 96;
    float acc = 0.f;
    for (int a = 0; a < EMBD; ++a) acc += z[n * EMBD + a] * t[(size_t)n * 768 + a * 96 + w];
    outs[i] += acc;
}

__global__ void contract_v_kernel(const float* __restrict__ z, const float* __restrict__ tv,
                                  float* __restrict__ outv) {
    int i = blockIdx.x * blockDim.x + threadIdx.x;
    if (i >= NN * 3 * 32) return;
    int ni = i / 32, w = i % 32, n = ni / 3;
    float acc = 0.f;
    for (int a = 0; a < EMBD; ++a) acc += z[n * EMBD + a] * tv[(size_t)ni * 256 + a * 32 + w];
    outv[i] += acc;
}

__global__ void gate_kernel(float* __restrict__ outs, float* __restrict__ outv) {
    int n = blockIdx.x * blockDim.x + threadIdx.x;
    if (n >= NN) return;
    for (int w = 0; w < 32; ++w) {
        float g = sigmoidf_(outs[(size_t)n * 96 + 64 + w]);
        outv[((size_t)n * 3 + 0) * 32 + w] *= g;
        outv[((size_t)n * 3 + 1) * 32 + w] *= g;
        outv[((size_t)n * 3 + 2) * 32 + w] *= g;
    }
    for (int c = 0; c < 64; ++c) {
        float v = outs[(size_t)n * 96 + c];
        outs[(size_t)n * 96 + c] = v * sigmoidf_(v);
    }
}

__global__ void integrate_kernel(const float* __restrict__ y, const float* __restrict__ yo,
                                 const float* __restrict__ gv, float* __restrict__ yn) {
    int i = blockIdx.x * blockDim.x + threadIdx.x;
    if (i >= NN * FEAT) return;
    int n = i / FEAT, f = i % FEAT;
    float ynew;
    if (f < NSC) ynew = yn[i]; // scalar part pre-filled by Wsis GEMM (ldc=160)
    else {
        int fv = f - NSC, v = fv / 3, c = fv % 3;
        ynew = gv[((size_t)n * 3 + c) * NVC + v];
    }
    yn[i] = 2.0f * y[i] - yo[i] + HSTEP * ynew;
}

__global__ void project_kernel(const float* __restrict__ y, const float* __restrict__ M,
                               float* __restrict__ xc) {
    int i = blockIdx.x * blockDim.x + threadIdx.x;
    if (i >= NN * 6) return;
    int n = i / 6, k = (i % 6) / 3, c = i % 3;
    float acc = 0.f;
    for (int v = 0; v < NVC; ++v) acc += y[(size_t)n * FEAT + NSC + v * 3 + c] * M[k * NVC + v];
    xc[i] = acc;
}

// ---------------------------------------------------------------------------
static inline int nblk(long n) { return (int)((n + 255) / 256); }

static void launch_gemm(const float* A, int lda, const float* B, int ldb,
                        float* C, int ldc, int Mrows, int K, int Ncol,
                        const float* bias, int mode, hipStream_t st) {
    int tilesN = Ncol / 16;
    int tiles = (Mrows / 16) * tilesN;
    int blocks = (tiles + 7) / 8;
    gemm_wmma_f32<<<blocks, 256, 0, st>>>(A, lda, B, ldb, C, ldc, K, bias, mode, tiles, tilesN);
}

extern "C" void kernel_launch(void* const* d_in, const int* in_sizes, int n_in,
                              void* d_out, int out_size, void* d_ws, size_t ws_size,
                              hipStream_t stream) {
    const float* x_in  = (const float*)d_in[0];
    const int*   attr  = (const int*)d_in[2];
    const int*   esrc  = (const int*)d_in[3];
    const int*   edst  = (const int*)d_in[4];
    const float* emb   = (const float*)d_in[5];
    const float* Mproj = (const float*)d_in[6];
    const float* W1s   = (const float*)d_in[7];
    const float* W1v   = (const float*)d_in[8];
    const float* Wr1   = (const float*)d_in[9];
    const float* br1   = (const float*)d_in[10];
    const float* Wr2   = (const float*)d_in[11];
    const float* W2s   = (const float*)d_in[12];
    const float* W2v   = (const float*)d_in[13];
    const float* Wscs  = (const float*)d_in[14];
    const float* Wscv  = (const float*)d_in[15];
    const float* Wsis  = (const float*)d_in[16];
    const float* Wsiv  = (const float*)d_in[17];

    float* ws = (float*)d_ws;
    size_t off = 0;
    float* yA  = ws + off; off += (size_t)NN * FEAT;
    float* yB  = ws + off; off += (size_t)NN * FEAT;
    float* yC  = ws + off; off += (size_t)NN * FEAT;
    float* z   = ws + off; off += (size_t)NN * EMBD;
    float* xc  = ws + off; off += (size_t)NN * 6;
    float* Mw  = ws + off; off += 64;
    float* s1  = ws + off; off += (size_t)NN * NSC;
    float* vt  = ws + off; off += (size_t)NN * 96;
    float* v1b = ws + off; off += (size_t)NN * 96;
    float* a0  = ws + off; off += (size_t)NN * NVC;
    float* av  = ws + off; off += (size_t)NN * 288;
    float* outs = ws + off; off += (size_t)NN * 96;
    float* outv = ws + off; off += (size_t)NN * 96;
    float* BIG  = ws + off; off += (size_t)NN * 768;

    // ---- preamble
    semiunitary_kernel<<<1, 32, 0, stream>>>(Mproj, Mw);
    zemb_kernel<<<nblk((long)NN * EMBD), 256, 0, stream>>>(attr, emb, z);
    (void)hipMemcpyAsync(xc, x_in, (size_t)NN * 6 * sizeof(float), hipMemcpyDeviceToDevice, stream);
    uplift_kernel<<<nblk((long)NN * FEAT), 256, 0, stream>>>(xc, Mw, yA);
    (void)hipMemcpyAsync(yB, yA, (size_t)NN * FEAT * sizeof(float), hipMemcpyDeviceToDevice, stream);

    float* y = yA; float* yo = yB; float* yn = yC;
    for (int l = 0; l < 2; ++l) {
        const float* w1s  = W1s  + (size_t)l * NSC * NSC;
        const float* w1v  = W1v  + (size_t)l * NVC * NVC;
        const float* wr1  = Wr1  + (size_t)l * NB * RH;
        const float* b1   = br1  + (size_t)l * RH;
        const float* wr2  = Wr2  + (size_t)l * RH * 128;
        const float* w2s  = W2s  + (size_t)l * NVC * 96;
        const float* w2v  = W2v  + (size_t)l * 96 * NVC;
        const float* wscs = Wscs + (size_t)l * NSC * EMBD * 96;
        const float* wscv = Wscv + (size_t)l * NVC * EMBD * NVC;
        const float* wsis = Wsis + (size_t)l * NSC * NSC;
        const float* wsiv = Wsiv + (size_t)l * NVC * NVC;

        // node pre-transforms
        launch_gemm(y, FEAT, w1s, NSC, s1, NSC, NN, NSC, NSC, nullptr, GOP_STORE, stream);
        repack_v_kernel<<<nblk((long)NN * 96), 256, 0, stream>>>(y, vt);
        launch_gemm(vt, NVC, w1v, NVC, v1b, NVC, NN * 3, NVC, NVC, nullptr, GOP_STORE, stream);

        // edge phase
        (void)hipMemsetAsync(a0, 0, (size_t)NN * NVC * sizeof(float), stream);
        (void)hipMemsetAsync(av, 0, (size_t)NN * 288 * sizeof(float), stream);
        edge_kernel<<<EE / 64, 128, EK_SMEM_FLOATS * sizeof(float), stream>>>(
            esrc, edst, xc, s1, v1b, wr1, b1, wr2, a0, av);

        // node post-transforms
        launch_gemm(a0, NVC, w2s, 96, outs, 96, NN, NVC, 96, nullptr, GOP_STORE, stream);
        launch_gemm(y, FEAT, wscs, 768, BIG, 768, NN, NSC, 768, nullptr, GOP_STORE, stream);
        contract_s_kernel<<<nblk((long)NN * 96), 256, 0, stream>>>(z, BIG, outs);
        launch_gemm(av, 96, w2v, NVC, outv, NVC, NN * 3, 96, NVC, nullptr, GOP_STORE, stream);
        launch_gemm(vt, NVC, wscv, 256, BIG, 256, NN * 3, NVC, 256, nullptr, GOP_STORE, stream);
        contract_v_kernel<<<nblk((long)NN * 96), 256, 0, stream>>>(z, BIG, outv);

        // gating + mixing
        gate_kernel<<<nblk(NN), 256, 0, stream>>>(outs, outv);
        launch_gemm(outs, 96, wsis, NSC, yn, FEAT, NN, NSC, NSC, nullptr, GOP_STORE, stream);
        launch_gemm(outv, NVC, wsiv, NVC, BIG, NVC, NN * 3, NVC, NVC, nullptr, GOP_STORE, stream);

        // integrate + re-project positions
        integrate_kernel<<<nblk((long)NN * FEAT), 256, 0, stream>>>(y, yo, BIG, yn);
        project_kernel<<<nblk((long)NN * 6), 256, 0, stream>>>(yn, Mw, xc);

        float* t = yo; yo = y; y = yn; yn = t;
    }
    (void)hipMemcpyAsync(d_out, xc, (size_t)NN * 6 * sizeof(float), hipMemcpyDeviceToDevice, stream);
}